// RoIHeads_31387620999322
// MI455X (gfx1250) — compile-verified
//
#include <hip/hip_runtime.h>
#include <stdint.h>

// ---------------- static problem configuration ----------------
#define Bn     8
#define Rn     4000
#define Cn     91
#define NCc    (Rn * (Cn - 1))      // 360000 candidates per image
#define Kn     2048                 // pre-NMS cap
#define NDETS  100
#define Mcap   4096                 // compaction capacity (sorted per image)
#define NBIN   4096                 // score histogram bins over [0,1]
#define SCORE_T 0.05f
#define NMS_T   0.5f
#define MIN_SZ  0.01f
#define DWDH_CLAMP 4.135166556742356f   // log(1000/16)
#define CLS_OFF 4096.0f                 // > max image coord (1216)

typedef unsigned __attribute__((ext_vector_type(4))) uint4v;
typedef unsigned __attribute__((ext_vector_type(8))) uint8v;

// ---- TDM helper: 1-D contiguous global->LDS copy of `elems` elements -------
// data_size_log2: 0=1B,1=2B,2=4B,3=8B ; elems must fit in 16-bit tile_dim0.
__device__ __forceinline__ void tdm_copy_1d(const void* gptr, unsigned ldsOff,
                                            unsigned elems, unsigned data_size_log2) {
  unsigned long long ga = (unsigned long long)gptr;
  uint4v g0;
  g0.x = 1u;                                                  // count=1 valid descriptor
  g0.y = ldsOff;                                              // lds_addr (bytes)
  g0.z = (unsigned)(ga & 0xFFFFFFFFull);                      // global_addr[31:0]
  g0.w = (unsigned)((ga >> 32) & 0x01FFFFFFull) | (2u << 30); // addr[56:32] | type=2
  uint8v g1;
  g1[0] = (data_size_log2 << 16);                             // data_size
  g1[1] = (elems & 0xFFFFu) << 16;                            // tensor_dim0[15:0]
  g1[2] = (elems >> 16) | (1u << 16);                         // tensor_dim0[31:16] | tensor_dim1=1
  g1[3] = (elems & 0xFFFFu) << 16;                            // tile_dim0
  g1[4] = 0u;                                                 // tile_dim1/2 unused
  g1[5] = elems;                                              // tensor_dim0_stride
  g1[6] = 0u; g1[7] = 0u;
  asm volatile("tensor_load_to_lds %0, %1" :: "s"(g0), "s"(g1) : "memory");
  __builtin_amdgcn_s_wait_tensorcnt(0);
}

// ---------------- shared decode helper (matches reference _decode + clip) ----
__device__ __forceinline__ void decode_clip(const float* __restrict__ reg,
                                            const float* __restrict__ prop,
                                            int b, int r, int c, float Hf, float Wf,
                                            float box[4]) {
  const float* p = prop + ((size_t)b * Rn + r) * 4;
  float px1 = p[0], py1 = p[1], px2 = p[2], py2 = p[3];
  float w = px2 - px1, h = py2 - py1;
  float cx = px1 + 0.5f * w, cy = py1 + 0.5f * h;
  const float* d = reg + (((size_t)b * Rn + r) * Cn + c) * 4;
  float dx = d[0] * 0.1f, dy = d[1] * 0.1f;
  float dw = fminf(d[2] * 0.2f, DWDH_CLAMP);
  float dh = fminf(d[3] * 0.2f, DWDH_CLAMP);
  float pcx = dx * w + cx, pcy = dy * h + cy;
  float pw = expf(dw) * w, ph = expf(dh) * h;
  box[0] = fminf(fmaxf(pcx - 0.5f * pw, 0.f), Wf);
  box[1] = fminf(fmaxf(pcy - 0.5f * ph, 0.f), Hf);
  box[2] = fminf(fmaxf(pcx + 0.5f * pw, 0.f), Wf);
  box[3] = fminf(fmaxf(pcy + 0.5f * ph, 0.f), Hf);
}

// ---------------- 0: zero per-launch state (harness does not re-zero ws) ----
__global__ void k_init(unsigned* __restrict__ hist, unsigned* __restrict__ cnt) {
  int g = blockIdx.x * blockDim.x + threadIdx.x;
  if (g < Bn * NBIN) hist[g] = 0u;
  if (g < Bn) cnt[g] = 0u;
}

// ---------------- 1: softmax, one wave32 per row of 91 logits ---------------
__global__ void k_softmax(const float* __restrict__ logits, float* __restrict__ scores) {
  int lane = threadIdx.x & 31;
  int row = blockIdx.x * (blockDim.x >> 5) + (threadIdx.x >> 5);
  if (row >= Bn * Rn) return;
  const float* in = logits + (size_t)row * Cn;
  float x0 = (lane      < Cn) ? in[lane]      : -3.0e38f;
  float x1 = (lane + 32 < Cn) ? in[lane + 32] : -3.0e38f;
  float x2 = (lane + 64 < Cn) ? in[lane + 64] : -3.0e38f;
  float m = fmaxf(x0, fmaxf(x1, x2));
  for (int d = 16; d; d >>= 1) m = fmaxf(m, __shfl_xor(m, d, 32));
  float e0 = (lane      < Cn) ? expf(x0 - m) : 0.f;
  float e1 = (lane + 32 < Cn) ? expf(x1 - m) : 0.f;
  float e2 = (lane + 64 < Cn) ? expf(x2 - m) : 0.f;
  float s = e0 + e1 + e2;
  for (int d = 16; d; d >>= 1) s += __shfl_xor(s, d, 32);
  float inv = 1.0f / s;
  float* out = scores + (size_t)row * Cn;
  if (lane      < Cn) out[lane]      = e0 * inv;
  if (lane + 32 < Cn) out[lane + 32] = e1 * inv;
  if (lane + 64 < Cn) out[lane + 64] = e2 * inv;
}

// ---------------- 2: decode all candidates, validity mask, score histogram --
__global__ void k_cand(const float* __restrict__ scores, const float* __restrict__ reg,
                       const float* __restrict__ prop, const int* __restrict__ shapes,
                       float* __restrict__ candS, unsigned* __restrict__ hist) {
  size_t gid = (size_t)blockIdx.x * blockDim.x + threadIdx.x;
  if (gid >= (size_t)Bn * NCc) return;
  int b = (int)(gid / NCc);
  int i = (int)(gid % NCc);
  int r = i / (Cn - 1);
  int c = i % (Cn - 1) + 1;                       // skip background class 0
  float Hf = (float)shapes[b * 2 + 0], Wf = (float)shapes[b * 2 + 1];
  float box[4];
  decode_clip(reg, prop, b, r, c, Hf, Wf, box);
  float s = scores[((size_t)b * Rn + r) * Cn + c];
  bool valid = (s > SCORE_T) && (box[2] - box[0] >= MIN_SZ) && (box[3] - box[1] >= MIN_SZ);
  candS[gid] = valid ? s : -1.0f;
  if (valid) {
    int bin = (int)(s * (float)NBIN);
    bin = bin > NBIN - 1 ? NBIN - 1 : bin;
    atomicAdd(&hist[b * NBIN + bin], 1u);
  }
}

// ---------------- 3: per-image threshold bin covering >= Kn entries ---------
__global__ void k_thresh(const unsigned* __restrict__ hist, int* __restrict__ tb) {
  int b = threadIdx.x;
  if (b >= Bn) return;
  unsigned cum = 0; int t = 0;
  for (int k = NBIN - 1; k >= 0; --k) {
    cum += hist[b * NBIN + k];
    if (cum >= Kn) { t = k; break; }
  }
  tb[b] = t;
}

// ---------------- 4: compact survivors as packed 64-bit sort keys -----------
__global__ void k_gather(const float* __restrict__ candS, const int* __restrict__ tb,
                         unsigned* __restrict__ cnt, unsigned long long* __restrict__ candList) {
  size_t gid = (size_t)blockIdx.x * blockDim.x + threadIdx.x;
  if (gid >= (size_t)Bn * NCc) return;
  int b = (int)(gid / NCc);
  int i = (int)(gid % NCc);
  float s = candS[gid];
  if (s <= 0.f) return;                            // invalid (stored -1)
  int bin = (int)(s * (float)NBIN);
  bin = bin > NBIN - 1 ? NBIN - 1 : bin;
  if (bin < tb[b]) return;
  unsigned pos = atomicAdd(&cnt[b], 1u);
  if (pos < Mcap) {
    unsigned u = __float_as_uint(s);
    unsigned ord = (u & 0x80000000u) ? ~u : (u | 0x80000000u);   // order-preserving
    // tie-break: lower candidate index first under descending sort -> pack ~i
    candList[(size_t)b * Mcap + pos] =
        ((unsigned long long)ord << 32) | (unsigned)(~(unsigned)i);
  }
}

// ---------------- 5: TDM-staged in-LDS bitonic sort (descending) ------------
__global__ void __launch_bounds__(1024) k_sort(const unsigned long long* __restrict__ candList,
                                               const unsigned* __restrict__ cnt,
                                               unsigned long long* __restrict__ sortedK) {
  __shared__ unsigned long long sk[Mcap];          // 32 KB
  int b = blockIdx.x;
  unsigned n = cnt[b]; if (n > Mcap) n = Mcap;
  if (threadIdx.x < 32) {
    // one TDM descriptor moves the whole 32KB key list global->LDS
    tdm_copy_1d(candList + (size_t)b * Mcap,
                (unsigned)(size_t)(&sk[0]), Mcap, /*8B*/3u);
  }
  __syncthreads();
  for (unsigned e = threadIdx.x; e < Mcap; e += blockDim.x)
    if (e >= n) sk[e] = 0ull;                      // stale tail must not sort
  __syncthreads();
  for (unsigned k = 2; k <= Mcap; k <<= 1) {
    for (unsigned jj = k >> 1; jj > 0; jj >>= 1) {
      for (unsigned e = threadIdx.x; e < Mcap; e += blockDim.x) {
        unsigned ix = e ^ jj;
        if (ix > e) {
          bool desc = ((e & k) == 0);
          unsigned long long a = sk[e], c2 = sk[ix];
          if (desc ? (a < c2) : (a > c2)) { sk[e] = c2; sk[ix] = a; }
        }
      }
      __syncthreads();
    }
  }
  for (unsigned e = threadIdx.x; e < Kn; e += blockDim.x)
    sortedK[(size_t)b * Kn + e] = sk[e];
}

// ---------------- 6: materialize top-K boxes (+class-offset NMS boxes) ------
__global__ void k_topk(const unsigned long long* __restrict__ sortedK,
                       const float* __restrict__ reg, const float* __restrict__ prop,
                       const int* __restrict__ shapes,
                       float* __restrict__ tkBox, float* __restrict__ tkNms,
                       float* __restrict__ tkScore, int* __restrict__ tkLabel,
                       int* __restrict__ tkValid) {
  int b = blockIdx.y;
  int j = blockIdx.x * blockDim.x + threadIdx.x;
  if (j >= Kn) return;
  size_t o1 = (size_t)b * Kn + j;
  size_t o4 = o1 * 4;
  unsigned long long key = sortedK[o1];
  if (key == 0ull) {                                // padding slot: inert box
    tkBox[o4] = tkBox[o4 + 1] = tkBox[o4 + 2] = tkBox[o4 + 3] = 0.f;
    tkNms[o4] = tkNms[o4 + 1] = tkNms[o4 + 2] = tkNms[o4 + 3] = 0.f;
    tkScore[o1] = 0.f; tkLabel[o1] = -1; tkValid[o1] = 0;
    return;
  }
  unsigned ord = (unsigned)(key >> 32);
  unsigned u = (ord & 0x80000000u) ? (ord & 0x7FFFFFFFu) : ~ord;
  float s = __uint_as_float(u);
  unsigned i = ~(unsigned)(key & 0xFFFFFFFFull);
  int r = (int)(i / (Cn - 1));
  int c = (int)(i % (Cn - 1)) + 1;
  float Hf = (float)shapes[b * 2 + 0], Wf = (float)shapes[b * 2 + 1];
  float box[4];
  decode_clip(reg, prop, b, r, c, Hf, Wf, box);
  float off = (float)c * CLS_OFF;                  // separates classes exactly
  tkBox[o4] = box[0]; tkBox[o4 + 1] = box[1]; tkBox[o4 + 2] = box[2]; tkBox[o4 + 3] = box[3];
  tkNms[o4] = box[0] + off; tkNms[o4 + 1] = box[1] + off;
  tkNms[o4 + 2] = box[2] + off; tkNms[o4 + 3] = box[3] + off;
  tkScore[o1] = s; tkLabel[o1] = c; tkValid[o1] = 1;
}

// ---------------- 7: IoU suppression bitmask; TDM stages 32KB box tile ------
__global__ void k_iou(const float* __restrict__ tkNms, unsigned* __restrict__ supmat) {
  __shared__ float4 lcol[Kn];                      // all 2048 boxes of image b (32 KB)
  int b = blockIdx.y;
  int i = blockIdx.x * blockDim.x + threadIdx.x;   // row index 0..2047

  if (threadIdx.x < 32) {
    // one TDM descriptor moves all Kn*4 fp32 box words global->LDS
    tdm_copy_1d(tkNms + (size_t)b * Kn * 4,
                (unsigned)(size_t)(&lcol[0]), Kn * 4, /*4B*/2u);
  }
  __syncthreads();

  float4 mb = lcol[i];
  float areaA = (mb.z - mb.x) * (mb.w - mb.y);
  for (int cb = (i >> 5); cb < Kn / 32; ++cb) {    // only j-words that can hold j > i
    unsigned word = 0u;
#pragma unroll 4
    for (int t = 0; t < 32; ++t) {
      float4 ob = lcol[cb * 32 + t];
      float areaB = (ob.z - ob.x) * (ob.w - ob.y);
      float lx = fmaxf(mb.x, ob.x), ly = fmaxf(mb.y, ob.y);
      float rx = fminf(mb.z, ob.z), ry = fminf(mb.w, ob.w);
      float iw = fmaxf(rx - lx, 0.f), ih = fmaxf(ry - ly, 0.f);
      float inter = iw * ih;
      float iou = inter / (areaA + areaB - inter + 1e-7f);
      word |= (iou > NMS_T ? 1u : 0u) << t;
    }
    supmat[((size_t)b * Kn + i) * (Kn / 32) + cb] = word;
  }
}

// ---------------- 8: greedy NMS scan; keep mask lives in VGPRs of one wave --
__global__ void k_nms(const int* __restrict__ tkValid, const unsigned* __restrict__ supmat,
                      unsigned* __restrict__ keepw) {
  int b = blockIdx.x;
  int lane = threadIdx.x;                          // wave32: 32 lanes, 2 words each
  unsigned kw0 = 0u, kw1 = 0u;
  for (int s2 = 0; s2 < 32; ++s2) {
    kw0 |= (tkValid[(size_t)b * Kn + lane * 32 + s2]        ? 1u : 0u) << s2;
    kw1 |= (tkValid[(size_t)b * Kn + (lane + 32) * 32 + s2] ? 1u : 0u) << s2;
  }
  for (int i = 0; i < Kn; ++i) {
    if (i + 1 < Kn) {                              // hide L2 latency of next row
      const unsigned* nrow = supmat + ((size_t)b * Kn + (i + 1)) * (Kn / 32);
      __builtin_prefetch(nrow + lane, 0, 1);       // -> global_prefetch_b8
      __builtin_prefetch(nrow + lane + 32, 0, 1);
    }
    int wi = i >> 5, bi = i & 31;
    unsigned src = (wi & 32) ? kw1 : kw0;
    unsigned alive = (__shfl(src, wi & 31, 32) >> bi) & 1u;  // uniform
    if (alive) {
      const unsigned* row = supmat + ((size_t)b * Kn + i) * (Kn / 32);
      if (lane > wi)                 kw0 &= ~row[lane];
      else if (lane == wi && bi < 31) kw0 &= ~(row[lane] & (0xFFFFFFFFu << (bi + 1)));
      int w1 = lane + 32;
      if (w1 > wi)                   kw1 &= ~row[w1];
      else if (w1 == wi && bi < 31)  kw1 &= ~(row[w1] & (0xFFFFFFFFu << (bi + 1)));
    }
  }
  keepw[b * (Kn / 32) + lane]      = kw0;
  keepw[b * (Kn / 32) + lane + 32] = kw1;
}

// ---------------- 9: final top-NDETS (already sorted desc; compact + pad) ---
__global__ void k_final(const float* __restrict__ tkBox, const float* __restrict__ tkScore,
                        const int* __restrict__ tkLabel, const unsigned* __restrict__ keepw,
                        float* __restrict__ outBS, int* __restrict__ outLab) {
  int b = blockIdx.x;
  if (threadIdx.x != 0) return;
  int n = 0;
  for (int j = 0; j < Kn && n < NDETS; ++j) {
    if ((keepw[b * (Kn / 32) + (j >> 5)] >> (j & 31)) & 1u) {
      float* o = outBS + ((size_t)b * NDETS + n) * 5;
      const float* bx = tkBox + ((size_t)b * Kn + j) * 4;
      o[0] = bx[0]; o[1] = bx[1]; o[2] = bx[2]; o[3] = bx[3];
      o[4] = tkScore[(size_t)b * Kn + j];
      outLab[b * NDETS + n] = tkLabel[(size_t)b * Kn + j];
      ++n;
    }
  }
  for (; n < NDETS; ++n) {
    float* o = outBS + ((size_t)b * NDETS + n) * 5;
    o[0] = o[1] = o[2] = o[3] = o[4] = 0.f;
    outLab[b * NDETS + n] = -1;
  }
}

// ---------------- host-side launch ------------------------------------------
extern "C" void kernel_launch(void* const* d_in, const int* in_sizes, int n_in,
                              void* d_out, int out_size, void* d_ws, size_t ws_size,
                              hipStream_t stream) {
  (void)in_sizes; (void)n_in; (void)out_size; (void)ws_size;
  const float* logits = (const float*)d_in[0];   // [B*R, C]
  const float* reg    = (const float*)d_in[1];   // [B*R, C*4]
  const float* prop   = (const float*)d_in[2];   // [B, R, 4]
  const int*   shapes = (const int*)d_in[3];     // [B, 2] (H, W)
  float* outBS  = (float*)d_out;                          // [B, 100, 5]
  int*   outLab = (int*)d_out + (size_t)Bn * NDETS * 5;   // [B, 100] int32

  size_t off = 0;
  auto take = [&](size_t bytes) -> char* {
    off = (off + 255) & ~(size_t)255;
    char* p = (char*)d_ws + off;
    off += bytes;
    return p;
  };
  float*    scores  = (float*)take(sizeof(float) * (size_t)Bn * Rn * Cn);
  float*    candS   = (float*)take(sizeof(float) * (size_t)Bn * NCc);
  unsigned* hist    = (unsigned*)take(sizeof(unsigned) * Bn * NBIN);
  int*      tb      = (int*)take(sizeof(int) * Bn);
  unsigned* cnt     = (unsigned*)take(sizeof(unsigned) * Bn);
  unsigned long long* candList = (unsigned long long*)take(8ull * Bn * Mcap);
  unsigned long long* sortedK  = (unsigned long long*)take(8ull * Bn * Kn);
  float*    tkBox   = (float*)take(sizeof(float) * (size_t)Bn * Kn * 4);
  float*    tkNms   = (float*)take(sizeof(float) * (size_t)Bn * Kn * 4);
  float*    tkScore = (float*)take(sizeof(float) * (size_t)Bn * Kn);
  int*      tkLabel = (int*)take(sizeof(int) * (size_t)Bn * Kn);
  int*      tkValid = (int*)take(sizeof(int) * (size_t)Bn * Kn);
  unsigned* supmat  = (unsigned*)take(sizeof(unsigned) * (size_t)Bn * Kn * (Kn / 32));
  unsigned* keepw   = (unsigned*)take(sizeof(unsigned) * Bn * (Kn / 32));

  size_t ncand = (size_t)Bn * NCc;
  k_init   <<<(Bn * NBIN + 255) / 256, 256, 0, stream>>>(hist, cnt);
  k_softmax<<<(Bn * Rn) / 8, 256, 0, stream>>>(logits, scores);
  k_cand   <<<(unsigned)((ncand + 255) / 256), 256, 0, stream>>>(scores, reg, prop, shapes, candS, hist);
  k_thresh <<<1, 32, 0, stream>>>(hist, tb);
  k_gather <<<(unsigned)((ncand + 255) / 256), 256, 0, stream>>>(candS, tb, cnt, candList);
  k_sort   <<<Bn, 1024, 0, stream>>>(candList, cnt, sortedK);
  k_topk   <<<dim3(Kn / 256, Bn), 256, 0, stream>>>(sortedK, reg, prop, shapes,
                                                    tkBox, tkNms, tkScore, tkLabel, tkValid);
  k_iou    <<<dim3(Kn / 256, Bn), 256, 0, stream>>>(tkNms, supmat);
  k_nms    <<<Bn, 32, 0, stream>>>(tkValid, supmat, keepw);
  k_final  <<<Bn, 1, 0, stream>>>(tkBox, tkScore, tkLabel, keepw, outBS, outLab);
}